// MedianPool2d_31645319037285
// MI455X (gfx1250) — compile-verified
//
#include <hip/hip_runtime.h>

// MedianPool2d (K=3, stride=1, width-only, replicate pad) on gfx1250.
// 65536 rows x 1024 fp32. Pure bandwidth problem: 512MB traffic, ~22us floor
// at 23.3 TB/s. One 256-thread block (8 wave32s) per row; async B128 copy of
// the row into LDS, med3 from LDS, NT B128 store.

#define W   1024
#define TPB 256   // 4 floats per thread

typedef float v4f __attribute__((ext_vector_type(4)));
typedef int   v4i __attribute__((ext_vector_type(4)));

// Address-space-qualified pointer types for the gfx1250 async-copy builtin.
typedef __attribute__((address_space(1))) v4i* gptr_v4i;   // global
typedef __attribute__((address_space(3))) v4i* lptr_v4i;   // LDS

#ifndef __has_builtin
#define __has_builtin(x) 0
#endif

#if __has_builtin(__builtin_amdgcn_global_load_async_to_lds_b128)
#define USE_ASYNC_LDS 1
#else
#define USE_ASYNC_LDS 0
#endif

__device__ __forceinline__ float med3(float a, float b, float c) {
#if __has_builtin(__builtin_amdgcn_fmed3f)
    return __builtin_amdgcn_fmed3f(a, b, c);   // v_med3_f32
#else
    float mn = fminf(a, b), mx = fmaxf(a, b);
    return fmaxf(mn, fminf(mx, c));
#endif
}

__global__ __launch_bounds__(TPB) void median3_w_kernel(
    const float* __restrict__ x, float* __restrict__ y)
{
    const int tid = threadIdx.x;
    const int c0  = tid << 2;                       // first column of this lane
    const long long row = blockIdx.x;
    const float* __restrict__ rowp = x + row * (long long)W;
    float*       __restrict__ outp = y + row * (long long)W;

    v4f v;
    float l, r;

#if USE_ASYNC_LDS
    // Stage the whole row in LDS via the gfx1250 async path: each lane moves
    // its 16B chunk with GLOBAL_LOAD_ASYNC_TO_LDS_B128 (tracked by ASYNCcnt).
    // Every global element is read exactly once; all overlapping neighbor
    // reads are then served from LDS.
    __shared__ __align__(16) float s[W];
    __builtin_amdgcn_global_load_async_to_lds_b128(
        (gptr_v4i)(rowp + c0),      // global source (AS1), 16B per lane
        (lptr_v4i)(&s[c0]),         // LDS destination (AS3)
        /*imm offset*/ 0, /*cpol*/ 0);
    asm volatile("s_wait_asynccnt 0" ::: "memory");
    __syncthreads();

    v = *(const v4f*)&s[c0];                        // ds_load_b128
    l = s[(c0 == 0)       ? 0       : c0 - 1];      // replicate-pad clamp
    r = s[(c0 + 4 >= W)   ? (W - 1) : c0 + 4];
#else
    // Fallback: direct B128 load + two clamped scalar neighbor loads
    // (neighbor lines are already resident in L0/L2 -> no extra HBM traffic).
    v = *(const v4f*)(rowp + c0);
    l = rowp[(c0 == 0)     ? 0       : c0 - 1];
    r = rowp[(c0 + 4 >= W) ? (W - 1) : c0 + 4];
#endif

    // Replicate-pad median-of-3: at the edges med3(x0,x0,x1)==x0 and
    // med3(xN-2,xN-1,xN-1)==xN-1, which matches the reference's pad exactly.
    v4f o;
    o.x = med3(l,   v.x, v.y);
    o.y = med3(v.x, v.y, v.z);
    o.z = med3(v.y, v.z, v.w);
    o.w = med3(v.z, v.w, r);

    // Output is write-once streamed data; NT hint keeps it out of L2.
    __builtin_nontemporal_store(o, (v4f*)(outp + c0));
}

extern "C" void kernel_launch(void* const* d_in, const int* in_sizes, int n_in,
                              void* d_out, int out_size, void* d_ws, size_t ws_size,
                              hipStream_t stream)
{
    const float* x = (const float*)d_in[0];
    float*       y = (float*)d_out;
    const int rows = in_sizes[0] / W;               // 32*1*2048 = 65536
    median3_w_kernel<<<dim3(rows), dim3(TPB), 0, stream>>>(x, y);
}